// LocalLinear_17016660427371
// MI455X (gfx1250) — compile-verified
//
#include <hip/hip_runtime.h>

// Sparse local-linear:  out[b,j] = bias[j] + sum_{e in segment(j)} w[e] * x[b, cols[e]]
// B = 32 (== wave32 lane count), rows sorted ascending.
//
// Pass 0: edge-parallel CSR build: row_ptr[j] = lower_bound(rows, j), j in [0, n_out].
// Pass 1: transpose x (B, N_IN) -> xT (N_IN, B) via CDNA5 async global->LDS
//         scatter (lane supplies coalesced global addr + transposed LDS addr),
//         s_wait_asynccnt, then coalesced write-out.
// Pass 2: one wave per output j (lane = batch). Edge metadata loaded
//         cooperatively (2 coalesced requests / 32 edges); full chunks are
//         fully unrolled so per-edge broadcast is a single v_readlane into
//         SGPRs (no DS traffic, scalar address math). Each edge costs exactly
//         one 128B L2-resident xT line. 32x33 LDS tile transposes results for
//         coalesced (B, N_OUT) stores with fused bias.

#define TILE 32  // == B == wave32 width

__global__ void rowptr_kernel(const int* __restrict__ rows,
                              int* __restrict__ row_ptr,
                              int nnz, int n_out) {
  const int e = blockIdx.x * blockDim.x + threadIdx.x;
  if (e >= nnz) return;
  const int r = rows[e];
  if (e == 0) {
    for (int j = 0; j <= r; ++j) row_ptr[j] = 0;           // lower_bound == 0
  } else {
    const int rp = rows[e - 1];
    for (int j = rp + 1; j <= r; ++j) row_ptr[j] = e;      // boundary fills gap
  }
  if (e == nnz - 1) {
    for (int j = r + 1; j <= n_out; ++j) row_ptr[j] = nnz; // trailing outputs
  }
}

__global__ __launch_bounds__(1024) void xpose_async_kernel(
    const float* __restrict__ x, float* __restrict__ xT, int n_in) {
  __shared__ float tile[TILE][TILE + 1];
  const int tx = threadIdx.x;
  const int ty = threadIdx.y;
  const int c0 = blockIdx.x * TILE;

  // Coalesced global read of x[ty][c0+tx], async-scattered into LDS
  // transposed: tile[tx][ty].  Pitch 33 -> conflict-free column scatter.
  const float* gsrc = x + (size_t)ty * (size_t)n_in + (size_t)(c0 + tx);
  unsigned lds_dst = (unsigned)(size_t)(&tile[tx][ty]);  // low 32 bits = LDS offset
  asm volatile("global_load_async_to_lds_b32 %0, %1, off"
               :: "v"(lds_dst), "v"(gsrc)
               : "memory");
#if __has_builtin(__builtin_amdgcn_s_wait_asynccnt)
  __builtin_amdgcn_s_wait_asynccnt(0);
#else
  asm volatile("s_wait_asynccnt 0" ::: "memory");
#endif
  __syncthreads();

  // xT[(c0+ty)][tx] = tile[ty][tx] = x[tx][c0+ty]; contiguous over tx.
  xT[(size_t)(c0 + ty) * TILE + tx] = tile[ty][tx];
}

__global__ __launch_bounds__(1024) void gather_segment_kernel(
    const float* __restrict__ xT, const float* __restrict__ w,
    const int* __restrict__ row_ptr, const int* __restrict__ cols,
    const float* __restrict__ bias, float* __restrict__ out, int n_out) {
  __shared__ float tile[TILE][TILE + 1];
  const int tx = threadIdx.x;          // batch lane b
  const int ty = threadIdx.y;          // output owned by this wave
  const int j0 = blockIdx.x * TILE;
  const int j  = j0 + ty;

  const int start = row_ptr[j];        // broadcast load (1 request/wave)
  const int end   = row_ptr[j + 1];

  float acc = 0.0f;
  int base = start;

  // Full 32-edge chunks: constant-lane readlane broadcast (SGPR, no DS ops).
  for (; base + TILE <= end; base += TILE) {
    const int      c_l = cols[base + tx];   // coalesced chunk load
    const unsigned w_l = __float_as_uint(w[base + tx]);
#pragma unroll
    for (int k = 0; k < TILE; ++k) {
      const int   c  = __builtin_amdgcn_readlane(c_l, k);   // scalar col
      const float wv = __uint_as_float(__builtin_amdgcn_readlane(w_l, k));
      acc = fmaf(wv, xT[c * TILE + tx], acc);  // one 128B L2 line per edge
    }
  }

  // Remainder (< 32 edges): guarded load + dynamic shuffle broadcast.
  if (base < end) {
    const int rem = end - base;
    int   c_l = 0;
    float w_l = 0.0f;
    if (tx < rem) {
      c_l = cols[base + tx];
      w_l = w[base + tx];
    }
    for (int k = 0; k < rem; ++k) {
      const int   c  = __shfl(c_l, k);
      const float wv = __shfl(w_l, k);
      acc = fmaf(wv, xT[c * TILE + tx], acc);
    }
  }

  // In-block transpose so the (B, N_OUT) stores are coalesced.
  tile[ty][tx] = acc;
  __syncthreads();
  const int jw = j0 + tx;
  out[(size_t)ty * (size_t)n_out + jw] = tile[tx][ty] + bias[jw];
}

extern "C" void kernel_launch(void* const* d_in, const int* in_sizes, int n_in,
                              void* d_out, int out_size, void* d_ws, size_t ws_size,
                              hipStream_t stream) {
  // setup_inputs order: x, weight, bias, rows, cols, N_out
  const float* x    = (const float*)d_in[0];
  const float* wgt  = (const float*)d_in[1];
  const float* bias = (const float*)d_in[2];
  const int*   rows = (const int*)d_in[3];   // int32 per harness convention
  const int*   cols = (const int*)d_in[4];
  float*       out  = (float*)d_out;

  const int NNZ   = in_sizes[1];
  const int N_OUT = in_sizes[2];
  const int B     = TILE;                    // 32 == out_size / N_OUT
  const int N_IN  = in_sizes[0] / B;

  float* xT      = (float*)d_ws;                         // N_IN*32 floats (8 MB)
  int*   row_ptr = (int*)(xT + (size_t)N_IN * TILE);     // n_out+1 ints (256 KB)

  hipLaunchKernelGGL(rowptr_kernel, dim3((NNZ + 255) / 256), dim3(256), 0, stream,
                     rows, row_ptr, NNZ, N_OUT);

  dim3 blk(TILE, TILE);
  hipLaunchKernelGGL(xpose_async_kernel, dim3(N_IN / TILE), blk, 0, stream,
                     x, xT, N_IN);
  hipLaunchKernelGGL(gather_segment_kernel, dim3(N_OUT / TILE), blk, 0, stream,
                     xT, wgt, row_ptr, cols, bias, out, N_OUT);
  (void)n_in; (void)out_size; (void)ws_size;
}